// CorrVolume1_4_79233556676724
// MI455X (gfx1250) — compile-verified
//
#include <hip/hip_runtime.h>

typedef __attribute__((ext_vector_type(2))) float v2f;
typedef __attribute__((ext_vector_type(8))) float v8f;

#define B_  8
#define H_  160
#define W_  320
#define C_  128
#define D_  48
#define WT  64      // output w-tile width per block
#define RW  112     // R tile width: 48 halo + 64
#define KC  64      // channels per LDS chunk (2 chunks over C=128)
#define SL  68      // padded LDS row stride in floats (bank = (4w+c)%64)

__global__ __launch_bounds__(256)
void corr_volume_wmma(const float* __restrict__ Lp,
                      const float* __restrict__ Rp,
                      float* __restrict__ out)
{
    __shared__ float sL[WT * SL];   // 17408 B
    __shared__ float sR[RW * SL];   // 30464 B  (total 47872 B)

    const int blk = blockIdx.x;             // B_*H_*(W_/WT) = 6400 blocks
    const int wt  = blk % (W_ / WT);        // 0..4
    const int bh  = blk / (W_ / WT);        // b*H_ + h
    const int w0  = wt * WT;
    const int h   = bh % H_;
    const int b   = bh / H_;

    const size_t rowBase = (size_t)bh * (size_t)(W_ * C_);

    const int tid  = threadIdx.x;
    const int lane = tid & 31;
    const int wave = tid >> 5;
    const int ti   = wave & 3;              // w sub-tile index 0..3
    const int tb   = (wave >> 2) * 2;       // band base t: 0 or 2 (wave owns t=tb, tb+1)

    v8f acc0 = {0.f,0.f,0.f,0.f,0.f,0.f,0.f,0.f};
    v8f acc1 = {0.f,0.f,0.f,0.f,0.f,0.f,0.f,0.f};

    // Fragment addressing (ISA 32-bit 16x4 A layout):
    //   lane<16:  VGPR0=A[m,k+0], VGPR1=A[m,k+1]; lane>=16: K offsets +2,+3
    const int am   = lane & 15;
    const int ak   = (lane >> 4) * 2;
    const int aoff = (ti * 16 + am) * SL + ak;
    // B[K=c, N=n] = R[w'(n), c];  w' tile base for t: w0 + 16*ti - 16*t,
    // LDS index base (relative to w0-48): 48 + 16*ti - 16*t
    const int rb0  = (48 + 16 * ti - 16 * tb + am) * SL + ak;
    const int rb1  = rb0 - 16 * SL;

    for (int kc = 0; kc < C_; kc += KC) {
        // ---- stage L chunk: 64 w x 64 c = 1024 float4, 4 per thread ----
        #pragma unroll
        for (int it = 0; it < 4; ++it) {
            int idx = tid + it * 256;
            int wl  = idx >> 4;                 // 0..63
            int c4  = (idx & 15) << 2;          // 0..60
            const float4 v = *(const float4*)(Lp + rowBase + (size_t)(w0 + wl) * C_ + kc + c4);
            *(float4*)(&sL[wl * SL + c4]) = v;
        }
        // ---- stage R chunk: 112 w' x 64 c = 1792 float4, 7 per thread (zero-pad w'<0) ----
        #pragma unroll
        for (int it = 0; it < 7; ++it) {
            int idx = tid + it * 256;
            int wl  = idx >> 4;                 // 0..111
            int c4  = (idx & 15) << 2;
            int wp  = w0 - 48 + wl;
            float4 v = make_float4(0.f, 0.f, 0.f, 0.f);
            if (wp >= 0)
                v = *(const float4*)(Rp + rowBase + (size_t)wp * C_ + kc + c4);
            *(float4*)(&sR[wl * SL + c4]) = v;
        }
        __syncthreads();

        // ---- 16 k-steps of fp32 WMMA per chunk; A shared by both band tiles ----
        #pragma unroll
        for (int k = 0; k < KC; k += 4) {
            v2f a, fb0, fb1;
            a.x   = sL[aoff + k];   a.y   = sL[aoff + k + 1];
            fb0.x = sR[rb0 + k];    fb0.y = sR[rb0 + k + 1];
            fb1.x = sR[rb1 + k];    fb1.y = sR[rb1 + k + 1];
            acc0 = __builtin_amdgcn_wmma_f32_16x16x4_f32(
                       false, a, false, fb0, (short)0, acc0, false, false);
            acc1 = __builtin_amdgcn_wmma_f32_16x16x4_f32(
                       false, a, false, fb1, (short)0, acc1, false, false);
        }
        __syncthreads();
    }

    // ---- scatter: D layout: VGPR v -> M = v + 8*(lane/16), N = lane%16 ----
    // d = m - n + 16*t ; write only valid band entries (each (d,w) exactly once)
    const int mhi = (lane >> 4) * 8;
    const int n   = lane & 15;
    #pragma unroll
    for (int v = 0; v < 8; ++v) {
        int m  = v + mhi;
        int w  = w0 + 16 * ti + m;
        int d0 = m - n + 16 * tb;
        int d1 = d0 + 16;
        size_t base = ((size_t)(b * D_) * H_ + (size_t)h) * W_ + (size_t)w;
        if (d0 >= 0 && d0 < D_) out[base + (size_t)d0 * (size_t)(H_ * W_)] = acc0[v];
        if (d1 >= 0 && d1 < D_) out[base + (size_t)d1 * (size_t)(H_ * W_)] = acc1[v];
    }
}

extern "C" void kernel_launch(void* const* d_in, const int* in_sizes, int n_in,
                              void* d_out, int out_size, void* d_ws, size_t ws_size,
                              hipStream_t stream) {
    const float* Lp = (const float*)d_in[0];
    const float* Rp = (const float*)d_in[1];
    float* out = (float*)d_out;

    dim3 grid(B_ * H_ * (W_ / WT));   // 8*160*5 = 6400 blocks
    dim3 block(256);                  // 8 wave32
    corr_volume_wmma<<<grid, block, 0, stream>>>(Lp, Rp, out);
}